// PureConv_5162550690706
// MI455X (gfx1250) — compile-verified
//
#include <hip/hip_runtime.h>

#define THREADS 256
#define EPB 512       // edges staged per block in the scatter kernel
#define DEG_EPB 1024  // edges per block in the degree kernel

typedef __attribute__((ext_vector_type(4))) unsigned v4u;
typedef __attribute__((ext_vector_type(8))) int      v8i;
typedef __attribute__((ext_vector_type(4))) int      v4i;

// ---------------------------------------------------------------------------
// TDM: 1-D tile DMA global->LDS via Tensor Data Mover (CDNA5 ISA ch.7/8).
// D# group0: count=1, lds_addr, global_addr[56:0], type=2.
// D# group1: data_size=4B, tensor_dim0=tensor_elems (OOB beyond -> zero-fill),
//            tensor_dim1=1, tile_dim0=tile_elems, tile_dim1=1, stride0=tile.
// Must be issued by ONE wave (TDM ignores EXEC); completion via TENSORcnt.
// This toolchain's builtin takes 6 args: (g0, g1, g2, g3, g_extra, cpol).
// ---------------------------------------------------------------------------
__device__ __forceinline__ void tdm_load_1d(unsigned lds_byte_addr,
                                            const void* gptr,
                                            unsigned tile_elems,
                                            unsigned tensor_elems) {
  unsigned long long ga = (unsigned long long)gptr;
  v4u g0;
  g0.x = 1u;                                                // count=1, user desc
  g0.y = lds_byte_addr;                                     // lds_addr
  g0.z = (unsigned)ga;                                      // global_addr[31:0]
  g0.w = (unsigned)((ga >> 32) & 0x01ffffffu) | (2u << 30); // [56:32] | type=2
  v8i g1;
  g1[0] = (int)(2u << 16);                                  // data_size=2 -> 4B
  g1[1] = (int)((tensor_elems & 0xffffu) << 16);            // tensor_dim0[15:0]
  g1[2] = (int)((tensor_elems >> 16) | (1u << 16));         // dim0[31:16], dim1=1
  g1[3] = (int)(tile_elems << 16);                          // tile_dim0
  g1[4] = 1;                                                // tile_dim1=1, tile_dim2=0
  g1[5] = (int)tile_elems;                                  // tensor_dim0_stride
  g1[6] = 0;
  g1[7] = 0;
  v4i z4 = {0, 0, 0, 0};
  v8i z8 = {0, 0, 0, 0, 0, 0, 0, 0};
#if __has_builtin(__builtin_amdgcn_tensor_load_to_lds)
  __builtin_amdgcn_tensor_load_to_lds(g0, g1, z4, z4, z8, 0);
#endif
}

__device__ __forceinline__ void wait_tensorcnt0() {
#if __has_builtin(__builtin_amdgcn_s_wait_tensorcnt)
  __builtin_amdgcn_s_wait_tensorcnt(0);
#else
  asm volatile("s_wait_tensorcnt 0" ::: "memory");
#endif
}

__device__ __forceinline__ void wait_asynccnt0() {
#if __has_builtin(__builtin_amdgcn_s_wait_asynccnt)
  __builtin_amdgcn_s_wait_asynccnt(0);
#else
  asm volatile("s_wait_asynccnt 0" ::: "memory");
#endif
}

// ---------------------------------------------------------------------------
// Kernel 1: degree histogram over edge_row.
// Full blocks async-stage 1024 indices into LDS (ASYNCcnt path, b128/lane),
// then histogram from LDS with int atomics.
// ---------------------------------------------------------------------------
__global__ void k_degree(const int* __restrict__ erow, int* __restrict__ deg,
                         int n_edges) {
  __shared__ int s_e[DEG_EPB];
  const int t = threadIdx.x;
  const long long base = (long long)blockIdx.x * DEG_EPB;
  if (base + DEG_EPB <= (long long)n_edges) {
    const int* g = erow + base + t * 4;
    unsigned l = (unsigned)(unsigned long long)(const void*)(s_e + t * 4);
    asm volatile("global_load_async_to_lds_b128 %0, %1, off"
                 :: "v"(l), "v"(g) : "memory");
    wait_asynccnt0();
    __syncthreads();
    int4 r = *(const int4*)(s_e + t * 4);
    atomicAdd(&deg[r.x], 1);
    atomicAdd(&deg[r.y], 1);
    atomicAdd(&deg[r.z], 1);
    atomicAdd(&deg[r.w], 1);
  } else {
    for (long long e = base + t; e < n_edges; e += THREADS)
      atomicAdd(&deg[erow[e]], 1);
  }
}

// ---------------------------------------------------------------------------
// Kernel 2: norm = rsqrt(1+deg); xs = norm*x (workspace); out = xs (residual).
// ---------------------------------------------------------------------------
__global__ void k_init(const float* __restrict__ x, const int* __restrict__ deg,
                       float* __restrict__ xs, float* __restrict__ out,
                       int n_nodes) {
  long long t = (long long)blockIdx.x * blockDim.x + threadIdx.x;
  long long i = t >> 3;
  if (i >= n_nodes) return;
  int fg = (int)(t & 7);
  float norm = rsqrtf(1.0f + (float)deg[i]);
  long long off = i * 32 + fg * 4;
  float4 v = *(const float4*)(x + off);
  v.x *= norm; v.y *= norm; v.z *= norm; v.w *= norm;
  *(float4*)(xs + off) = v;
  *(float4*)(out + off) = v;
}

// ---------------------------------------------------------------------------
// Kernel 3: scatter-add  out[row[e]] += xs[col[e]]
// Row/col index tiles DMA'd into LDS by the Tensor Data Mover (wave 0 issues,
// TENSORcnt wait, barrier). TDM zero-fills reads past tensor_dim0, so tail
// blocks need no special staging path. Gathers are 128B-contiguous per edge
// (8 lanes x float4); scatters are contiguous global_atomic_add_f32.
// ---------------------------------------------------------------------------
__global__ void k_scatter(const int* __restrict__ erow,
                          const int* __restrict__ ecol,
                          const float* __restrict__ xs,
                          float* __restrict__ out, int n_edges) {
  __shared__ int s_row[EPB];
  __shared__ int s_col[EPB];
  const int t = threadIdx.x;
  const long long base = (long long)blockIdx.x * EPB;

  if (t < 32) {  // wave 0 only: TDM issues regardless of EXEC, gate by branch
    unsigned remaining = (unsigned)((long long)n_edges - base);
    unsigned lr = (unsigned)(unsigned long long)(const void*)s_row;
    unsigned lc = (unsigned)(unsigned long long)(const void*)s_col;
    tdm_load_1d(lr, erow + base, EPB, remaining);
    tdm_load_1d(lc, ecol + base, EPB, remaining);
    wait_tensorcnt0();
  }
  __syncthreads();

  // 512 edges * 8 lanes-per-edge / 256 threads = 16 passes.
  // Lanes 0..7 of a wave share an edge -> gather/scatter hit one 128B line.
  if (base + EPB <= (long long)n_edges) {
#pragma unroll
    for (int it = 0; it < (EPB * 8) / THREADS; ++it) {
      int w = it * THREADS + t;
      int le = w >> 3;
      int fg = w & 7;
      int r = s_row[le];
      int c = s_col[le];
      float4 v = *(const float4*)(xs + (long long)c * 32 + fg * 4);
      float* dst = out + (long long)r * 32 + fg * 4;
      atomicAdd(dst + 0, v.x);
      atomicAdd(dst + 1, v.y);
      atomicAdd(dst + 2, v.z);
      atomicAdd(dst + 3, v.w);
    }
  } else {
    for (int it = 0; it < (EPB * 8) / THREADS; ++it) {
      int w = it * THREADS + t;
      int le = w >> 3;
      long long e = base + le;
      if (e < n_edges) {
        int fg = w & 7;
        int r = s_row[le];
        int c = s_col[le];
        float4 v = *(const float4*)(xs + (long long)c * 32 + fg * 4);
        float* dst = out + (long long)r * 32 + fg * 4;
        atomicAdd(dst + 0, v.x);
        atomicAdd(dst + 1, v.y);
        atomicAdd(dst + 2, v.z);
        atomicAdd(dst + 3, v.w);
      }
    }
  }
}

// ---------------------------------------------------------------------------
// Kernel 4: out *= rsqrt(1+deg)   (second scaling, in place)
// ---------------------------------------------------------------------------
__global__ void k_finalize(float* __restrict__ out, const int* __restrict__ deg,
                           int n_nodes) {
  long long t = (long long)blockIdx.x * blockDim.x + threadIdx.x;
  long long i = t >> 3;
  if (i >= n_nodes) return;
  int fg = (int)(t & 7);
  float norm = rsqrtf(1.0f + (float)deg[i]);
  long long off = i * 32 + fg * 4;
  float4 v = *(float4*)(out + off);
  v.x *= norm; v.y *= norm; v.z *= norm; v.w *= norm;
  *(float4*)(out + off) = v;
}

// ---------------------------------------------------------------------------
extern "C" void kernel_launch(void* const* d_in, const int* in_sizes, int n_in,
                              void* d_out, int out_size, void* d_ws,
                              size_t ws_size, hipStream_t stream) {
  const float* x  = (const float*)d_in[0];
  const int* erow = (const int*)d_in[1];
  const int* ecol = (const int*)d_in[2];
  const int n_edges = in_sizes[1];
  const int n_nodes = in_sizes[0] / 32;  // D_FEAT = 32
  float* out = (float*)d_out;

  // Workspace layout: [deg: n_nodes ints][xs: n_nodes*32 floats]
  int* deg = (int*)d_ws;
  size_t deg_bytes = (((size_t)n_nodes * sizeof(int)) + 255) & ~(size_t)255;
  float* xs = (float*)((char*)d_ws + deg_bytes);

  (void)hipMemsetAsync(deg, 0, (size_t)n_nodes * sizeof(int), stream);

  int g_deg = (n_edges + DEG_EPB - 1) / DEG_EPB;
  k_degree<<<g_deg, THREADS, 0, stream>>>(erow, deg, n_edges);

  long long node_threads = (long long)n_nodes * 8;
  int g_node = (int)((node_threads + THREADS - 1) / THREADS);
  k_init<<<g_node, THREADS, 0, stream>>>(x, deg, xs, out, n_nodes);

  int g_sc = (n_edges + EPB - 1) / EPB;
  k_scatter<<<g_sc, THREADS, 0, stream>>>(erow, ecol, xs, out, n_edges);

  k_finalize<<<g_node, THREADS, 0, stream>>>(out, deg, n_nodes);
}